// CNN_OTAM_85761906967065
// MI455X (gfx1250) — compile-verified
//
#include <hip/hip_runtime.h>

// Problem constants (from reference setup_inputs)
#define QN 512
#define SN 25
#define LN 16
#define DN 2048
#define NWAY 5
#define LBDA_F 0.1f
#define INV_LBDA_F 10.0f
#define EPS_F 0.01f

typedef __attribute__((ext_vector_type(2))) float v2f;
typedef __attribute__((ext_vector_type(8))) float v8f;

// ---------------------------------------------------------------------------
// Kernel 1: row L2 norms. One wave (32 lanes) per row, coalesced strided reads,
// butterfly reduction via __shfl_xor (wave32 on gfx1250).
// ---------------------------------------------------------------------------
__global__ void otam_row_norms(const float* __restrict__ x, float* __restrict__ out,
                               int rows, int d) {
    int wave = (int)((blockIdx.x * blockDim.x + threadIdx.x) >> 5);
    int lane = (int)(threadIdx.x & 31);
    if (wave >= rows) return;
    const float* row = x + (size_t)wave * d;
    float s = 0.0f;
    for (int j = lane; j < d; j += 32) {
        float v = row[j];
        s = fmaf(v, v, s);
    }
    #pragma unroll
    for (int off = 16; off > 0; off >>= 1) s += __shfl_xor(s, off, 32);
    if (lane == 0) out[wave] = sqrtf(s);
}

// ---------------------------------------------------------------------------
// Kernel 2: WMMA fp32 GEMM + cosine-distance epilogue.
// C[m,n] = sum_k tf[m,k]*sf[n,k]  (NT GEMM), M=Q*L=8192, N=S*L=400, K=2048.
//
// Tiling: one wave = 2 query clips (2 x 16 rows) x 5 support clips (5 x 16
// cols) = 10 accumulator tiles (80 VGPRs). Per k-group of 8 we batch-load
// 14 b64 fragments into distinct registers (one s_clause, overlapped waits),
// then issue 20 v_wmma_f32_16x16x4_f32. B fragments are reused by both row
// tiles, halving L2 traffic vs 1-row-tile tiling (~11.7 FLOP/B from cache).
//
// fp32 16x4 A fragment (ISA 7.12.2): lane&15 = M row; VGPR0 holds K = k+2*half,
// VGPR1 holds K = k+2*half+1  =>  per-lane contiguous float2 at row[k+2*half].
// B (4x16, K x N) mirrors this with N = lane&15, so identical addressing.
// C/D fragment: VGPR i -> row l = i + 8*(lane>>4), col = lane&15.
// ---------------------------------------------------------------------------
__global__ void otam_gemm_dists(const float* __restrict__ tf,
                                const float* __restrict__ sf,
                                const float* __restrict__ ntf,
                                const float* __restrict__ nsf,
                                float* __restrict__ dists) {
    int wave = (int)((blockIdx.x * blockDim.x + threadIdx.x) >> 5);
    int lane = (int)(threadIdx.x & 31);
    // 256 row-pair groups x 5 s-groups = 1280 waves
    int gq = wave / 5;       // query pair index: q0 = 2*gq, q1 = 2*gq+1
    int sg = wave % 5;       // support group: s = sg*5 + t
    if (gq >= QN / 2) return;

    int half = lane >> 4;    // 0: lanes 0-15, 1: lanes 16-31
    int r    = lane & 15;    // M index for A, N index for B, col for C

    const float* aptr0 = tf + (size_t)((2 * gq + 0) * LN + r) * DN;
    const float* aptr1 = tf + (size_t)((2 * gq + 1) * LN + r) * DN;
    const float* brow[5];
    #pragma unroll
    for (int t = 0; t < 5; ++t) {
        int scol = (sg * 5 + t) * LN + r;
        brow[t] = sf + (size_t)scol * DN;
    }

    v8f acc0[5] = {};
    v8f acc1[5] = {};

    int koff = 2 * half;
    for (int k = 0; k < DN; k += 8) {
        // ---- batch-load all 14 fragments for this k-group ----
        v2f a00 = *(const v2f*)(aptr0 + k + koff);
        v2f a01 = *(const v2f*)(aptr0 + k + 4 + koff);
        v2f a10 = *(const v2f*)(aptr1 + k + koff);
        v2f a11 = *(const v2f*)(aptr1 + k + 4 + koff);
        v2f b0[5], b1[5];
        #pragma unroll
        for (int t = 0; t < 5; ++t) {
            b0[t] = *(const v2f*)(brow[t] + k + koff);
            b1[t] = *(const v2f*)(brow[t] + k + 4 + koff);
        }
        // ---- 20 WMMAs, each B fragment reused by both row tiles ----
        #pragma unroll
        for (int t = 0; t < 5; ++t) {
            acc0[t] = __builtin_amdgcn_wmma_f32_16x16x4_f32(
                false, a00, false, b0[t], (short)0, acc0[t], false, false);
            acc1[t] = __builtin_amdgcn_wmma_f32_16x16x4_f32(
                false, a10, false, b0[t], (short)0, acc1[t], false, false);
        }
        #pragma unroll
        for (int t = 0; t < 5; ++t) {
            acc0[t] = __builtin_amdgcn_wmma_f32_16x16x4_f32(
                false, a01, false, b1[t], (short)0, acc0[t], false, false);
            acc1[t] = __builtin_amdgcn_wmma_f32_16x16x4_f32(
                false, a11, false, b1[t], (short)0, acc1[t], false, false);
        }
    }

    // Epilogue: 1 - cos_sim, scattered into [Q,S,L,L]
    #pragma unroll
    for (int rt = 0; rt < 2; ++rt) {
        int q = 2 * gq + rt;
        float nt[8];
        #pragma unroll
        for (int i = 0; i < 8; ++i) nt[i] = ntf[q * LN + (i + 8 * half)];
        #pragma unroll
        for (int t = 0; t < 5; ++t) {
            int s = sg * 5 + t;
            float ns = nsf[s * LN + r];
            size_t base = (((size_t)q * SN + s) * LN) * LN + r;
            #pragma unroll
            for (int i = 0; i < 8; ++i) {
                int l = i + 8 * half;
                float num = rt ? acc1[t][i] : acc0[t][i];
                float den = fmaf(nt[i], ns, EPS_F);
                dists[base + (size_t)l * LN] = 1.0f - num / den;
            }
        }
    }
}

// ---------------------------------------------------------------------------
// Kernel 3: OTAM soft-min DP. One lane per (q,s) pair (12800 lanes).
// Faithful to reference: padded M = L+2 = 18, column 0 stays 0, row 0 is a
// running sum, interior cells soft-min over {prev[m-1], new[m-1]}, boundary
// columns m==1 and m==17 additionally include prev[m]. Two passes (dists and
// dists^T), summed. Stable: val = d + mn - lambda*log(sum exp((mn-x)/lambda)).
// m-loop fully unrolled so prev[18] stays in VGPRs.
// ---------------------------------------------------------------------------
__global__ void otam_dp(const float* __restrict__ dists, float* __restrict__ cum) {
    int t = (int)(blockIdx.x * blockDim.x + threadIdx.x);
    if (t >= QN * SN) return;
    const float* blk = dists + (size_t)t * (LN * LN);

    float total = 0.0f;
    for (int pass = 0; pass < 2; ++pass) {
        int rs = pass ? 1 : LN;   // row stride within the tile
        int cs = pass ? LN : 1;   // col stride within the tile

        float prev[18];
        prev[0] = 0.0f;
        #pragma unroll
        for (int m = 1; m <= 16; ++m)
            prev[m] = prev[m - 1] + blk[(m - 1) * cs];
        prev[17] = prev[16];      // padded d[0][17] = 0

        for (int l = 1; l < LN; ++l) {
            const float* row = blk + l * rs;
            float pm1 = prev[0];  // old prev[m-1]
            float cur = 0.0f;     // new[m-1]; new[0] = 0
            #pragma unroll
            for (int m = 1; m <= 17; ++m) {
                float pm = prev[m];                       // old prev[m]
                float dm = (m <= 16) ? row[(m - 1) * cs] : 0.0f;
                bool boundary = (m == 1) || (m == 17);
                float mn = fminf(pm1, cur);
                if (boundary) mn = fminf(mn, pm);
                float ssum = __expf((mn - pm1) * INV_LBDA_F)
                           + __expf((mn - cur) * INV_LBDA_F);
                if (boundary) ssum += __expf((mn - pm) * INV_LBDA_F);
                float val = dm + mn - LBDA_F * __logf(ssum);
                prev[m] = val;
                pm1 = pm;
                cur = val;
            }
        }
        total += prev[17];
    }
    cum[t] = total;
}

// ---------------------------------------------------------------------------
// Kernel 4: per-class mean over support and negate. One thread per query.
// ---------------------------------------------------------------------------
__global__ void otam_class_reduce(const float* __restrict__ cum,
                                  const int* __restrict__ labels,
                                  float* __restrict__ out) {
    int q = (int)(blockIdx.x * blockDim.x + threadIdx.x);
    if (q >= QN) return;
    float sums[NWAY] = {0.f, 0.f, 0.f, 0.f, 0.f};
    float cnts[NWAY] = {0.f, 0.f, 0.f, 0.f, 0.f};
    for (int s = 0; s < SN; ++s) {
        int c = labels[s];
        float v = cum[q * SN + s];
        #pragma unroll
        for (int k = 0; k < NWAY; ++k) {
            if (c == k) { sums[k] += v; cnts[k] += 1.0f; }
        }
    }
    #pragma unroll
    for (int c = 0; c < NWAY; ++c)
        out[q * NWAY + c] = -(sums[c] / cnts[c]);
}

// ---------------------------------------------------------------------------
// Launch
// ---------------------------------------------------------------------------
extern "C" void kernel_launch(void* const* d_in, const int* in_sizes, int n_in,
                              void* d_out, int out_size, void* d_ws, size_t ws_size,
                              hipStream_t stream) {
    const float* tf     = (const float*)d_in[0];  // [Q, L, D]
    const float* sf     = (const float*)d_in[1];  // [S, L, D]
    const int*   labels = (const int*)d_in[2];    // [S]
    float* out = (float*)d_out;                   // [Q, NWAY]

    // Workspace layout (floats)
    float* ws    = (float*)d_ws;
    float* ntf   = ws;                                // Q*L      = 8192
    float* nsf   = ws + 8192;                         // S*L      = 400 (pad to 512)
    float* dists = ws + 8192 + 512;                   // Q*S*L*L  = 3,276,800
    float* cum   = dists + (size_t)QN * SN * LN * LN; // Q*S      = 12,800
    (void)ws_size; (void)in_sizes; (void)n_in; (void)out_size;

    // 1) row norms: one wave per row, 8 waves per 256-thread block
    {
        int rowsT = QN * LN;  // 8192
        int waves_per_block = 256 / 32;
        int blocksT = (rowsT + waves_per_block - 1) / waves_per_block;
        otam_row_norms<<<blocksT, 256, 0, stream>>>(tf, ntf, rowsT, DN);
        int rowsS = SN * LN;  // 400
        int blocksS = (rowsS + waves_per_block - 1) / waves_per_block;
        otam_row_norms<<<blocksS, 256, 0, stream>>>(sf, nsf, rowsS, DN);
    }

    // 2) WMMA GEMM + cosine-distance epilogue:
    //    (QN/2) row-pair groups x 5 s-groups = 1280 waves, 8 waves/block
    {
        int total_waves = (QN / 2) * 5;   // 1280
        int blocks = total_waves / 8;     // 160
        otam_gemm_dists<<<blocks, 256, 0, stream>>>(tf, sf, ntf, nsf, dists);
    }

    // 3) OTAM DP: one lane per (q,s)
    {
        int n = QN * SN;  // 12800
        otam_dp<<<(n + 255) / 256, 256, 0, stream>>>(dists, cum);
    }

    // 4) class-mean reduction
    {
        otam_class_reduce<<<(QN + 255) / 256, 256, 0, stream>>>(cum, labels, out);
    }
}